// ParticleQuantumEncoder_33380485825091
// MI455X (gfx1250) — compile-verified
//
#include <hip/hip_runtime.h>
#include <hip/hip_bf16.h>

typedef __attribute__((ext_vector_type(2))) float v2f;
typedef __attribute__((ext_vector_type(8))) float v8f;

// ============ slot-bit (register-register) gate helpers, 32 slots ==========
template<int Q>
__device__ __forceinline__ void rx_s(float (&re)[32], float (&im)[32], float c, float s) {
  #pragma unroll
  for (int r = 0; r < 32; ++r) {
    if ((r >> Q) & 1) continue;
    const int r1 = r | (1 << Q);
    float x0r = re[r], x0i = im[r], x1r = re[r1], x1i = im[r1];
    re[r]  = c * x0r + s * x1i;  im[r]  = c * x0i - s * x1r;
    re[r1] = c * x1r + s * x0i;  im[r1] = c * x1i - s * x0r;
  }
}
template<int Q>
__device__ __forceinline__ void ry_s(float (&re)[32], float (&im)[32], float c, float s) {
  #pragma unroll
  for (int r = 0; r < 32; ++r) {
    if ((r >> Q) & 1) continue;
    const int r1 = r | (1 << Q);
    float x0r = re[r], x0i = im[r], x1r = re[r1], x1i = im[r1];
    re[r]  = c * x0r - s * x1r;  im[r]  = c * x0i - s * x1i;
    re[r1] = s * x0r + c * x1r;  im[r1] = s * x0i + c * x1i;
  }
}
template<int Q>
__device__ __forceinline__ void rz_s(float (&re)[32], float (&im)[32], float c, float s) {
  #pragma unroll
  for (int r = 0; r < 32; ++r) {
    const float t = ((r >> Q) & 1) ? -s : s;
    float mr = re[r], mi = im[r];
    re[r] = c * mr + t * mi;
    im[r] = c * mi - t * mr;
  }
}
template<int QC, int QT>   // CNOT: both bits are slot bits -> register rename
__device__ __forceinline__ void cnot_ss(float (&re)[32], float (&im)[32]) {
  #pragma unroll
  for (int r = 0; r < 32; ++r) {
    if ((((r >> QC) & 1) == 1) && (((r >> QT) & 1) == 0)) {
      const int r1 = r | (1 << QT);
      float tr = re[r], ti = im[r];
      re[r] = re[r1]; im[r] = im[r1];
      re[r1] = tr;    im[r1] = ti;
    }
  }
}
// CNOT(bit5,bit4) in L1 layout: control = lane bit 0, target = slot bit 4.
// Per-lane conditional register swap (v_cndmask), no data movement.
__device__ __forceinline__ void cnot_l1_54(float (&re)[32], float (&im)[32], int lane) {
  const bool ctrl = (lane & 1) != 0;
  #pragma unroll
  for (int r = 0; r < 16; ++r) {
    const int r1 = r + 16;
    float ar = re[r], br = re[r1], ai = im[r], bi = im[r1];
    re[r]  = ctrl ? br : ar;   re[r1] = ctrl ? ar : br;
    im[r]  = ctrl ? bi : ai;   im[r1] = ctrl ? ai : bi;
  }
}

// ============ LDS layout transpose (L0 <-> L1), XOR-swizzled, 8KB/wave ======
// L0: idx = r*32 + lane.  L1: idx = lane*32 + r.
// offset(idx) = (idx & ~31) | ((idx ^ (idx>>5)) & 31)   [float2 units]
// -> both phases are LDS bank-conflict free.  Same-wave LDS is in-order.
template<bool TOL1>
__device__ __forceinline__ void ltr(float (&re)[32], float (&im)[32],
                                    float* __restrict__ wb, int lane) {
  float2* p = (float2*)wb;
  #pragma unroll
  for (int r = 0; r < 32; ++r) {
    const int xo = lane ^ r;
    const int so = TOL1 ? (r * 32 + xo) : (lane * 32 + xo);
    p[so] = make_float2(re[r], im[r]);
  }
  __asm__ __volatile__("" ::: "memory");   // keep compiler from reordering ld/st
  #pragma unroll
  for (int r = 0; r < 32; ++r) {
    const int xo = lane ^ r;
    const int lo = TOL1 ? (lane * 32 + xo) : (r * 32 + xo);
    const float2 v = p[lo];
    re[r] = v.x; im[r] = v.y;
  }
  __asm__ __volatile__("" ::: "memory");
}

// ============ jet kernel: 1 wave = 1 batch element, 8 waves/block ==========
__global__ __launch_bounds__(256, 1)
void pqe_jet_kernel(const float* __restrict__ x, const float* __restrict__ jw,
                    float* __restrict__ out, int nb) {
  extern __shared__ float smem[];            // 8 * 2048 floats = 64 KB
  const int lane = threadIdx.x & 31;
  const int wave = threadIdx.x >> 5;
  const int b = blockIdx.x * 8 + wave;
  float* wb = smem + wave * 2048;

  if (b < nb) {
    const float* row = x + (size_t)b * 56;
    float re[32], im[32];

    // ---- layer 1 (RX,RY,RZ on every wire applied to |0..0>) is a product
    // state: build ten 2-vectors, then materialize directly in L1 layout
    // (idx bits[9:5]=lane -> wires 0..4, bits[4:0]=slot -> wires 5..9).
    float v0r[10], v0i[10], v1r[10], v1i[10];
    #pragma unroll
    for (int k = 0; k < 10; ++k) {
      float cx, sx, cy, sy, cz, sz;
      __sincosf(0.5f * row[36 + k], &sx, &cx);   // RX(eta)
      __sincosf(0.5f * row[26 + k], &sy, &cy);   // RY(pt)
      __sincosf(0.5f * row[46 + k], &sz, &cz);   // RZ(phi)
      float a0r = cx, a0i = 0.0f, a1r = 0.0f, a1i = -sx;        // RX|0>
      float b0r = cy * a0r - sy * a1r, b0i = cy * a0i - sy * a1i; // RY
      float b1r = sy * a0r + cy * a1r, b1i = sy * a0i + cy * a1i;
      v0r[k] = cz * b0r + sz * b0i;  v0i[k] = cz * b0i - sz * b0r; // RZ diag
      v1r[k] = cz * b1r - sz * b1i;  v1i[k] = cz * b1i + sz * b1r;
    }
    float plr = 1.0f, pli = 0.0f;              // lane factor: wires 0..4
    #pragma unroll
    for (int k = 0; k < 5; ++k) {
      const int bit = (lane >> (4 - k)) & 1;
      float fr = bit ? v1r[k] : v0r[k];
      float fi = bit ? v1i[k] : v0i[k];
      float nr = plr * fr - pli * fi, ni = plr * fi + pli * fr;
      plr = nr; pli = ni;
    }
    re[0] = plr; im[0] = pli;                  // prefix-product tree: wires 5..9
    #pragma unroll
    for (int j = 0; j < 5; ++j) {
      const int k = 5 + j;
      const int n = 1 << j;
      #pragma unroll
      for (int t = n - 1; t >= 0; --t) {
        float ar = re[t], ai = im[t];
        re[2*t+1] = ar * v1r[k] - ai * v1i[k];  im[2*t+1] = ar * v1i[k] + ai * v1r[k];
        re[2*t]   = ar * v0r[k] - ai * v0i[k];  im[2*t]   = ar * v0i[k] + ai * v0r[k];
      }
    }
    // state now in L1

    float c, s;
#define CNOT_HI() cnot_ss<4,3>(re,im); cnot_ss<3,2>(re,im); cnot_ss<2,1>(re,im); cnot_ss<1,0>(re,im);
#define CNOT_LO() cnot_l1_54(re,im,lane); cnot_ss<4,3>(re,im); cnot_ss<3,2>(re,im); cnot_ss<2,1>(re,im); cnot_ss<1,0>(re,im);
#define RX_LO(D) /* wires 5..9 in L1: slot bit 9-k */ \
    __sincosf(0.5f*jw[(D)*10+5], &s,&c); rx_s<4>(re,im,c,s); \
    __sincosf(0.5f*jw[(D)*10+6], &s,&c); rx_s<3>(re,im,c,s); \
    __sincosf(0.5f*jw[(D)*10+7], &s,&c); rx_s<2>(re,im,c,s); \
    __sincosf(0.5f*jw[(D)*10+8], &s,&c); rx_s<1>(re,im,c,s); \
    __sincosf(0.5f*jw[(D)*10+9], &s,&c); rx_s<0>(re,im,c,s);
#define RX_HI(D) /* wires 0..4 in L0: slot bit 4-k */ \
    __sincosf(0.5f*jw[(D)*10+0], &s,&c); rx_s<4>(re,im,c,s); \
    __sincosf(0.5f*jw[(D)*10+1], &s,&c); rx_s<3>(re,im,c,s); \
    __sincosf(0.5f*jw[(D)*10+2], &s,&c); rx_s<2>(re,im,c,s); \
    __sincosf(0.5f*jw[(D)*10+3], &s,&c); rx_s<1>(re,im,c,s); \
    __sincosf(0.5f*jw[(D)*10+4], &s,&c); rx_s<0>(re,im,c,s);

    // initial CNOT chain: (9,8)..(6,5) in L0; (5,4)..(1,0) in L1
    ltr<false>(re, im, wb, lane);  CNOT_HI()
    ltr<true >(re, im, wb, lane);  CNOT_LO()
    // depth layer 0 (RX on all wires commute within the layer)
    RX_LO(0)
    ltr<false>(re, im, wb, lane);  RX_HI(0)  CNOT_HI()
    ltr<true >(re, im, wb, lane);  CNOT_LO()
    // depth layer 1
    RX_LO(1)
    ltr<false>(re, im, wb, lane);  RX_HI(1)  CNOT_HI()
    ltr<true >(re, im, wb, lane);  CNOT_LO()
#undef RX_HI
#undef RX_LO
#undef CNOT_LO
#undef CNOT_HI

    // probabilities, L1 layout: idx = lane*32 + r  (contiguous per lane)
    #pragma unroll
    for (int r = 0; r < 32; ++r)
      wb[lane * 32 + r] = re[r] * re[r] + im[r] * im[r];
  } else {
    #pragma unroll
    for (int r = 0; r < 32; ++r) wb[lane * 32 + r] = 0.0f;
  }

  __syncthreads();

  // out[b][9+w] = sum_k probs[b][k] * (1 - 2*bit(9-w,k)):
  // 8x16x1024 f32 GEMM tile via V_WMMA_F32_16X16X4_F32, K stepped by 4.
  if (wave == 0) {
    const int col  = lane & 15;               // N (wire) for B/D; M row for A
    const int half = lane >> 4;               // K-half select (f32 A/B layout)
    const float* arow = smem + (size_t)(col & 7) * 2048;  // rows 8..15 dup 0..7
    const int   sh    = (col < 10) ? (9 - col) : 0;
    const float valid = (col < 10) ? 1.0f : 0.0f;

    v8f acc = {};
    for (int kk = 0; kk < 1024; kk += 4) {
      const int k0 = kk + half * 2;
      v2f a; a.x = arow[k0]; a.y = arow[k0 + 1];
      v2f bm;
      bm.x = valid * (1.0f - 2.0f * (float)((k0 >> sh) & 1));
      bm.y = valid * (1.0f - 2.0f * (float)(((k0 + 1) >> sh) & 1));
      acc = __builtin_amdgcn_wmma_f32_16x16x4_f32(false, a, false, bm,
                                                  (short)0, acc, false, false);
    }
    if (half == 0 && col < 10) {              // D rows 0..7 live in lanes 0..15
      #pragma unroll
      for (int v = 0; v < 8; ++v) {
        const int bb = blockIdx.x * 8 + v;
        if (bb < nb) out[(size_t)bb * 19 + 9 + col] = acc[v];
      }
    }
  }
}

// ============ met / ele / mu: 1 thread = 1 batch element ===================
__device__ __forceinline__ void rx16(float (&ar)[16], float (&ai)[16], int p, float c, float s) {
  #pragma unroll
  for (int i = 0; i < 16; ++i) {
    if ((i >> p) & 1) continue;
    const int j = i | (1 << p);
    float x0r = ar[i], x0i = ai[i], x1r = ar[j], x1i = ai[j];
    ar[i] = c * x0r + s * x1i;  ai[i] = c * x0i - s * x1r;
    ar[j] = c * x1r + s * x0i;  ai[j] = c * x1i - s * x0r;
  }
}
__device__ __forceinline__ void ry16(float (&ar)[16], float (&ai)[16], int p, float c, float s) {
  #pragma unroll
  for (int i = 0; i < 16; ++i) {
    if ((i >> p) & 1) continue;
    const int j = i | (1 << p);
    float x0r = ar[i], x0i = ai[i], x1r = ar[j], x1i = ai[j];
    ar[i] = c * x0r - s * x1r;  ai[i] = c * x0i - s * x1i;
    ar[j] = s * x0r + c * x1r;  ai[j] = s * x0i + c * x1i;
  }
}
__device__ __forceinline__ void rz16(float (&ar)[16], float (&ai)[16], int p, float c, float s) {
  #pragma unroll
  for (int i = 0; i < 16; ++i) {
    const float t = ((i >> p) & 1) ? -s : s;
    float mr = ar[i], mi = ai[i];
    ar[i] = c * mr + t * mi;
    ai[i] = c * mi - t * mr;
  }
}
__device__ __forceinline__ void cn16(float (&ar)[16], float (&ai)[16], int pc, int pt) {
  #pragma unroll
  for (int i = 0; i < 16; ++i) {
    if ((((i >> pc) & 1) == 1) && (((i >> pt) & 1) == 0)) {
      const int j = i | (1 << pt);
      float tr = ar[i], ti = ai[i];
      ar[i] = ar[j]; ai[i] = ai[j];
      ar[j] = tr;    ai[j] = ti;
    }
  }
}

__device__ __forceinline__ void run4(const float* pt, const float* eta, const float* phi,
                                     const float* w, float* outp) {
  float ar[16], ai[16];
  #pragma unroll
  for (int i = 0; i < 16; ++i) { ar[i] = 0.0f; ai[i] = 0.0f; }
  ar[0] = 1.0f;
  float c, s;
  #pragma unroll
  for (int k = 0; k < 4; ++k) {
    const int p = 3 - k;
    __sincosf(0.5f * eta[k], &s, &c); rx16(ar, ai, p, c, s);
    __sincosf(0.5f * pt[k],  &s, &c); ry16(ar, ai, p, c, s);
    __sincosf(0.5f * phi[k], &s, &c); rz16(ar, ai, p, c, s);
  }
  #pragma unroll
  for (int k = 0; k < 3; ++k) cn16(ar, ai, 3 - k, 2 - k);
  #pragma unroll
  for (int d = 0; d < 2; ++d) {
    #pragma unroll
    for (int k = 0; k < 4; ++k) {
      __sincosf(0.5f * w[d * 4 + k], &s, &c);
      rx16(ar, ai, 3 - k, c, s);
    }
    #pragma unroll
    for (int k = 0; k < 3; ++k) cn16(ar, ai, 3 - k, 2 - k);
  }
  float ev[4] = {0.0f, 0.0f, 0.0f, 0.0f};
  #pragma unroll
  for (int i = 0; i < 16; ++i) {
    const float pb = ar[i] * ar[i] + ai[i] * ai[i];
    #pragma unroll
    for (int wq = 0; wq < 4; ++wq)
      ev[wq] += ((i >> (3 - wq)) & 1) ? -pb : pb;
  }
  #pragma unroll
  for (int wq = 0; wq < 4; ++wq) outp[wq] = ev[wq];
}

__global__ void pqe_small_kernel(const float* __restrict__ x, const float* __restrict__ mw,
                                 const float* __restrict__ ew, const float* __restrict__ uw,
                                 float* __restrict__ out, int nb) {
  const int b = blockIdx.x * blockDim.x + threadIdx.x;
  if (b >= nb) return;
  const float* row = x + (size_t)b * 56;
  float* orow = out + (size_t)b * 19;

  {  // MET: RY(pt), RZ(phi), RX(w0), RX(w1), <Z>
    float a0r, a0i = 0.0f, a1r, a1i = 0.0f;
    float c, s;
    __sincosf(0.5f * row[0], &s, &c);
    a0r = c; a1r = s;
    __sincosf(0.5f * row[1], &s, &c);
    { float t0r = c * a0r + s * a0i, t0i = c * a0i - s * a0r;
      float t1r = c * a1r - s * a1i, t1i = c * a1i + s * a1r;
      a0r = t0r; a0i = t0i; a1r = t1r; a1i = t1i; }
    #pragma unroll
    for (int d = 0; d < 2; ++d) {
      __sincosf(0.5f * mw[d], &s, &c);
      float n0r = c * a0r + s * a1i, n0i = c * a0i - s * a1r;
      float n1r = c * a1r + s * a0i, n1i = c * a1i - s * a0r;
      a0r = n0r; a0i = n0i; a1r = n1r; a1i = n1i;
    }
    orow[0] = (a0r * a0r + a0i * a0i) - (a1r * a1r + a1i * a1i);
  }
  run4(row + 2,  row + 6,  row + 10, ew, orow + 1);   // ele
  run4(row + 14, row + 18, row + 22, uw, orow + 5);   // mu
}

extern "C" void kernel_launch(void* const* d_in, const int* in_sizes, int n_in,
                              void* d_out, int out_size, void* d_ws, size_t ws_size,
                              hipStream_t stream) {
  const float* x  = (const float*)d_in[0];
  const float* mw = (const float*)d_in[1];
  const float* ew = (const float*)d_in[2];
  const float* uw = (const float*)d_in[3];
  const float* jw = (const float*)d_in[4];
  float* out = (float*)d_out;
  const int nb = in_sizes[0] / 56;              // 4096

  pqe_small_kernel<<<(nb + 255) / 256, 256, 0, stream>>>(x, mw, ew, uw, out, nb);

  const int jblocks = (nb + 7) / 8;
  pqe_jet_kernel<<<jblocks, 256, 8 * 2048 * sizeof(float), stream>>>(x, jw, out, nb);
}